// GAE_29059748725634
// MI455X (gfx1250) — compile-verified
//
#include <hip/hip_runtime.h>
#include <hip/hip_bf16.h>

#define N_NODES 100000
#define N_EDGES 1600000
#define IN_F    128
#define HID_F   128
#define ENC_F   64
#define NCLS    5
#define BN_EPS  1e-5f

typedef __attribute__((ext_vector_type(2))) float v2f;
typedef __attribute__((ext_vector_type(8))) float v8f;

// ---------------------------------------------------------------------------
// Edge scatter: agg[dst[e]][:] += hin[src[e]][:]   (ncols = 128 or 64)
// One float4 (4 cols) per thread; a wave covers whole rows -> coalesced gather,
// contiguous atomic bursts into the L2-resident agg table.
// ---------------------------------------------------------------------------
__global__ __launch_bounds__(256) void scatter_add_kernel(
    const float* __restrict__ hin, const int* __restrict__ src,
    const int* __restrict__ dst, float* __restrict__ agg, int ncols)
{
    long tid = (long)blockIdx.x * blockDim.x + threadIdx.x;
    int groups = ncols >> 2;
    long total = (long)N_EDGES * groups;
    if (tid >= total) return;
    int e  = (int)(tid / groups);
    int cg = (int)(tid % groups);
    int s = src[e], d = dst[e];
    const float4 v = *reinterpret_cast<const float4*>(hin + (long)s * ncols + cg * 4);
    float* p = agg + (long)d * ncols + cg * 4;
    atomicAdd(p + 0, v.x);
    atomicAdd(p + 1, v.y);
    atomicAdd(p + 2, v.z);
    atomicAdd(p + 3, v.w);
}

// ---------------------------------------------------------------------------
// out = relu(A @ W + bias), fused per-column sum / sum-of-squares for BatchNorm.
// One wave per 16x16 output tile, K-loop of v_wmma_f32_16x16x4_f32 (fp32 WMMA,
// matches reference numerics). A:[N,K] row-major, W:[K,ncols] row-major.
// ---------------------------------------------------------------------------
__global__ __launch_bounds__(256) void gemm_bias_relu_stats(
    const float* __restrict__ A, const float* __restrict__ W,
    const float* __restrict__ bias, float* __restrict__ out,
    float* __restrict__ colsum, float* __restrict__ colsq,
    int K, int ncols)
{
    int wave  = (blockIdx.x * blockDim.x + threadIdx.x) >> 5;
    int lane  = threadIdx.x & 31;
    int tilesN = ncols >> 4;
    int m0 = (wave / tilesN) << 4;
    int n0 = (wave % tilesN) << 4;
    if (m0 >= N_NODES) return;                 // wave-uniform: EXEC stays all-ones

    int half = lane >> 4;                      // 0: K+0/1   1: K+2/3
    int l    = lane & 15;

    // A fragment (16x4 f32): lane l holds row m0+l, pair at k + 2*half
    const float* arow  = A + (long)(m0 + l) * K + 2 * half;
    // B fragment (4x16 f32): lane l holds col n0+l, pair at k + 2*half (stride ncols)
    const float* wbase = W + (long)(2 * half) * ncols + n0 + l;

    v8f acc = {};
    for (int k = 0; k < K; k += 4) {
        v2f a; a.x = arow[k];                  a.y = arow[k + 1];
        v2f b; b.x = wbase[(long)k * ncols];   b.y = wbase[(long)(k + 1) * ncols];
        acc = __builtin_amdgcn_wmma_f32_16x16x4_f32(
            false, a, false, b, (short)0, acc, false, false);
    }

    // Epilogue: C/D layout -> VGPR r is row m0 + 8*half + r, column n0 + l
    int   n  = n0 + l;
    float bn = bias[n];
    float s = 0.f, sq = 0.f;
    int rowbase = m0 + 8 * half;
#pragma unroll
    for (int r = 0; r < 8; ++r) {
        float v = acc[r] + bn;
        v = v > 0.f ? v : 0.f;
        out[(long)(rowbase + r) * ncols + n] = v;
        s += v; sq += v * v;
    }
    // lanes l and l+16 hold the same column -> fold halves, 2 atomics per column
    s  += __shfl_down(s, 16, 32);
    sq += __shfl_down(sq, 16, 32);
    if (half == 0) {
        atomicAdd(&colsum[n], s);
        atomicAdd(&colsq[n], sq);
    }
}

// ---------------------------------------------------------------------------
// BatchNorm apply (train mode, biased variance): elementwise, float4 per thread.
// ---------------------------------------------------------------------------
__global__ __launch_bounds__(256) void bn_apply_kernel(
    const float* __restrict__ h, const float* __restrict__ colsum,
    const float* __restrict__ colsq, const float* __restrict__ gamma,
    const float* __restrict__ beta, float* __restrict__ out, int ncols)
{
    long tid = ((long)blockIdx.x * blockDim.x + threadIdx.x) * 4;
    long total = (long)N_NODES * ncols;
    if (tid >= total) return;
    const float inv_n = 1.0f / (float)N_NODES;
    float4 v = *reinterpret_cast<const float4*>(h + tid);
    float r[4] = {v.x, v.y, v.z, v.w};
    float o[4];
    int nbase = (int)(tid % ncols);            // ncols % 4 == 0 -> no wrap in group
#pragma unroll
    for (int i = 0; i < 4; ++i) {
        int n = nbase + i;
        float mean = colsum[n] * inv_n;
        float var  = colsq[n] * inv_n - mean * mean;
        float scl  = gamma[n] * rsqrtf(var + BN_EPS);
        o[i] = (r[i] - mean) * scl + beta[n];
    }
    float4 ov = {o[0], o[1], o[2], o[3]};
    *reinterpret_cast<float4*>(out + tid) = ov;
}

// ---------------------------------------------------------------------------
// Decoder: per edge, p = enc2[src]*enc2[dst]; logits = p@Wd + bd; softmax.
// Wd (64x5) indices are wave-uniform -> scalar-cached loads.
// ---------------------------------------------------------------------------
__global__ __launch_bounds__(256) void decoder_kernel(
    const float* __restrict__ enc2, const int* __restrict__ src,
    const int* __restrict__ dst, const float* __restrict__ Wd,
    const float* __restrict__ bd, const int* __restrict__ ew,
    float* __restrict__ pred, float* __restrict__ gt)
{
    int e = blockIdx.x * blockDim.x + threadIdx.x;
    if (e >= N_EDGES) return;
    const float* ps = enc2 + (long)src[e] * ENC_F;
    const float* pd = enc2 + (long)dst[e] * ENC_F;
    float logit[NCLS];
#pragma unroll
    for (int c = 0; c < NCLS; ++c) logit[c] = bd[c];
#pragma unroll 4
    for (int i = 0; i < ENC_F; i += 4) {
        float4 a = *reinterpret_cast<const float4*>(ps + i);
        float4 b = *reinterpret_cast<const float4*>(pd + i);
        float p0 = a.x * b.x, p1 = a.y * b.y, p2 = a.z * b.z, p3 = a.w * b.w;
#pragma unroll
        for (int c = 0; c < NCLS; ++c) {
            logit[c] += p0 * Wd[(i + 0) * NCLS + c] + p1 * Wd[(i + 1) * NCLS + c]
                      + p2 * Wd[(i + 2) * NCLS + c] + p3 * Wd[(i + 3) * NCLS + c];
        }
    }
    float m = logit[0];
#pragma unroll
    for (int c = 1; c < NCLS; ++c) m = fmaxf(m, logit[c]);
    float ex[NCLS];
    float sum = 0.f;
#pragma unroll
    for (int c = 0; c < NCLS; ++c) { ex[c] = __expf(logit[c] - m); sum += ex[c]; }
    float inv = 1.0f / sum;
    float* po = pred + (long)e * NCLS;
#pragma unroll
    for (int c = 0; c < NCLS; ++c) po[c] = ex[c] * inv;
    gt[e] = (float)(ew[e] - 1);
}

// ---------------------------------------------------------------------------
extern "C" void kernel_launch(void* const* d_in, const int* in_sizes, int n_in,
                              void* d_out, int out_size, void* d_ws, size_t ws_size,
                              hipStream_t stream)
{
    (void)in_sizes; (void)n_in; (void)out_size; (void)ws_size;

    const float* x   = (const float*)d_in[0];
    const float* W1  = (const float*)d_in[1];
    const float* b1  = (const float*)d_in[2];
    const float* g1  = (const float*)d_in[3];
    const float* be1 = (const float*)d_in[4];
    const float* W2  = (const float*)d_in[5];
    const float* b2  = (const float*)d_in[6];
    const float* g2  = (const float*)d_in[7];
    const float* be2 = (const float*)d_in[8];
    const float* Wd  = (const float*)d_in[9];
    const float* bd  = (const float*)d_in[10];
    const int*   src = (const int*)d_in[11];
    const int*   dst = (const int*)d_in[12];
    const int*   ew  = (const int*)d_in[13];

    // workspace layout
    float* agg1  = (float*)d_ws;                           // N*128
    float* h1    = agg1 + (size_t)N_NODES * HID_F;         // N*128
    float* agg2  = h1   + (size_t)N_NODES * HID_F;         // N*64
    float* stats = agg2 + (size_t)N_NODES * ENC_F;         // 384 floats
    float* cs1 = stats;
    float* cq1 = cs1 + HID_F;
    float* cs2 = cq1 + HID_F;
    float* cq2 = cs2 + ENC_F;
    float* h2  = agg1;                                     // reuse after layer 1

    // output layout: predicted [E,5] | ground_truth [E] | enc2 [N,64]
    float* pred = (float*)d_out;
    float* gt   = pred + (size_t)N_EDGES * NCLS;
    float* enc2 = gt + (size_t)N_EDGES;

    // zero accumulators + BN stats (graph-capturable memset nodes)
    hipMemsetAsync(agg1, 0, (size_t)N_NODES * HID_F * sizeof(float), stream);
    hipMemsetAsync(agg2, 0, ((size_t)N_NODES * ENC_F + 2 * HID_F + 2 * ENC_F) * sizeof(float), stream);

    // ---- layer 1 -----------------------------------------------------------
    {
        long total = (long)N_EDGES * (IN_F / 4);
        int blocks = (int)((total + 255) / 256);
        scatter_add_kernel<<<blocks, 256, 0, stream>>>(x, src, dst, agg1, IN_F);
    }
    {
        int waves = (N_NODES / 16) * (HID_F / 16);         // 50000, divides evenly
        gemm_bias_relu_stats<<<waves / 8, 256, 0, stream>>>(
            agg1, W1, b1, h1, cs1, cq1, IN_F, HID_F);
    }
    {
        long total = (long)N_NODES * HID_F / 4;
        bn_apply_kernel<<<(int)((total + 255) / 256), 256, 0, stream>>>(
            h1, cs1, cq1, g1, be1, h1, HID_F);             // in-place -> enc1
    }

    // ---- layer 2 -----------------------------------------------------------
    {
        long total = (long)N_EDGES * (HID_F / 4);
        int blocks = (int)((total + 255) / 256);
        scatter_add_kernel<<<blocks, 256, 0, stream>>>(h1, src, dst, agg2, HID_F);
    }
    {
        int waves = (N_NODES / 16) * (ENC_F / 16);         // 25000, divides evenly
        gemm_bias_relu_stats<<<waves / 8, 256, 0, stream>>>(
            agg2, W2, b2, h2, cs2, cq2, HID_F, ENC_F);
    }
    {
        long total = (long)N_NODES * ENC_F / 4;
        bn_apply_kernel<<<(int)((total + 255) / 256), 256, 0, stream>>>(
            h2, cs2, cq2, g2, be2, enc2, ENC_F);           // -> d_out enc2 slot
    }

    // ---- decoder -----------------------------------------------------------
    {
        int blocks = (N_EDGES + 255) / 256;
        decoder_kernel<<<blocks, 256, 0, stream>>>(enc2, src, dst, Wd, bd, ew, pred, gt);
    }
}